// FourStageFeatureExtractor_74285754351966
// MI455X (gfx1250) — compile-verified
//
#include <hip/hip_runtime.h>
#include <hip/hip_bf16.h>
#include <math.h>

// ---------------------------------------------------------------------------
// MI455X (gfx1250) implementation of the 4-stage VSS feature extractor.
// All projections run as fp32 WMMA GEMMs (V_WMMA_F32_16X16X4_F32): workload is
// bandwidth/latency bound (~4 GFLOP-MAC vs ~150MB traffic @ 23.3 TB/s), so we
// keep full fp32 precision while still using the matrix pipe.
// Channel-last (B*L x C) everywhere -> every projection is an NT GEMM with the
// (out,in) weights used directly. Small weight tensors are zero-padded so that
// every GEMM has M%64==0, N%16==0, K%4==0 -> branch-free inner loop.
// ---------------------------------------------------------------------------

typedef __attribute__((ext_vector_type(2))) float v2f;
typedef __attribute__((ext_vector_type(8))) float v8f;

#define BATCH 4
#define HH 32
#define WW 32
#define LL (HH * WW)          // 1024
#define BL (BATCH * LL)       // 4096
#define NSTATE 16
#define C2P 48                // padded x_proj output width (dtr+32 -> 48)
#define MTILE 64              // rows of D per wave (4 x 16x16 sub-tiles)

// ------------------------- WMMA fp32 NT GEMM -------------------------------
// D[m,n] = sum_k A[m,k] * Bw[n,k]  (+ bias[n]) (+ add[m,n]) (+ relu)
// Requirements (guaranteed by caller): M%64==0, N%16==0, K%4==0, lds even.
// One wave computes a 64x16 tile: 4 accumulators sharing one B operand.
__global__ void wmma_gemm_nt(const float* __restrict__ A, int lda,
                             const float* __restrict__ Bw, int ldb,
                             float* __restrict__ D, int ldd,
                             const float* __restrict__ bias,
                             const float* __restrict__ addsrc, int ldadd,
                             int M, int N, int K, int relu) {
  const int lane = threadIdx.x & 31;
  const int wave = threadIdx.x >> 5;
  const int tilesN = N >> 4;
  const int tilesM = M / MTILE;
  const int tid = blockIdx.x * (blockDim.x >> 5) + wave;
  if (tid >= tilesM * tilesN) return;      // wave-uniform exit (EXEC stays full)
  const int tm = tid / tilesN;
  const int tn = tid - tm * tilesN;
  const int m0 = tm * MTILE, n0 = tn << 4;

  const int half = lane >> 4;              // K pair select: {0,1} or {2,3}
  const int idx = lane & 15;               // A row / B column within tile
  const int koff = 2 * half;

  const float* __restrict__ ar0 = A + (size_t)(m0 + idx) * lda + koff;
  const float* __restrict__ ar1 = ar0 + (size_t)16 * lda;
  const float* __restrict__ ar2 = ar0 + (size_t)32 * lda;
  const float* __restrict__ ar3 = ar0 + (size_t)48 * lda;
  const float* __restrict__ br  = Bw + (size_t)(n0 + idx) * ldb + koff;

  v8f acc0 = {}, acc1 = {}, acc2 = {}, acc3 = {};
  v2f a0 = *(const v2f*)ar0;
  v2f a1 = *(const v2f*)ar1;
  v2f a2 = *(const v2f*)ar2;
  v2f a3 = *(const v2f*)ar3;
  v2f bb = *(const v2f*)br;

  for (int k = 4; k < K; k += 4) {
    // software pipeline: issue next-k loads before consuming current operands
    v2f na0 = *(const v2f*)(ar0 + k);
    v2f na1 = *(const v2f*)(ar1 + k);
    v2f na2 = *(const v2f*)(ar2 + k);
    v2f na3 = *(const v2f*)(ar3 + k);
    v2f nb  = *(const v2f*)(br + k);
    __builtin_prefetch(ar0 + k + 64, 0, 1);          // global_prefetch_b8
    acc0 = __builtin_amdgcn_wmma_f32_16x16x4_f32(false, a0, false, bb,
                                                 (short)0, acc0, false, false);
    acc1 = __builtin_amdgcn_wmma_f32_16x16x4_f32(false, a1, false, bb,
                                                 (short)0, acc1, false, false);
    acc2 = __builtin_amdgcn_wmma_f32_16x16x4_f32(false, a2, false, bb,
                                                 (short)0, acc2, false, false);
    acc3 = __builtin_amdgcn_wmma_f32_16x16x4_f32(false, a3, false, bb,
                                                 (short)0, acc3, false, false);
    a0 = na0; a1 = na1; a2 = na2; a3 = na3; bb = nb;
  }
  acc0 = __builtin_amdgcn_wmma_f32_16x16x4_f32(false, a0, false, bb,
                                               (short)0, acc0, false, false);
  acc1 = __builtin_amdgcn_wmma_f32_16x16x4_f32(false, a1, false, bb,
                                               (short)0, acc1, false, false);
  acc2 = __builtin_amdgcn_wmma_f32_16x16x4_f32(false, a2, false, bb,
                                               (short)0, acc2, false, false);
  acc3 = __builtin_amdgcn_wmma_f32_16x16x4_f32(false, a3, false, bb,
                                               (short)0, acc3, false, false);

  // ---- epilogue: D layout VGPR r -> row (r + 8*half), lane idx -> column ----
  const float bv = bias ? bias[n0 + idx] : 0.f;
  const int col = n0 + idx;
  auto store = [&](v8f acc, int mb) {
#pragma unroll
    for (int r = 0; r < 8; ++r) {
      const size_t m = (size_t)(mb + r + 8 * half);
      float v = acc[r] + bv;
      if (addsrc) v += addsrc[m * ldadd + col];
      if (relu) v = fmaxf(v, 0.f);
      D[m * ldd + col] = v;
    }
  };
  store(acc0, m0);
  store(acc1, m0 + 16);
  store(acc2, m0 + 32);
  store(acc3, m0 + 48);
}

// ----------------------- LayerNorm (wave per row) --------------------------
// Optional fused gate: y = LN(x)*g+b, then *= silu(Z[row, c]).
__global__ void layernorm_gate(const float* __restrict__ X, int ldx,
                               const float* __restrict__ g,
                               const float* __restrict__ b,
                               const float* __restrict__ Z, int ldz,
                               float* __restrict__ Y, int ldy,
                               int rows, int C) {
  const int w = (int)((blockIdx.x * blockDim.x + threadIdx.x) >> 5);
  const int lane = threadIdx.x & 31;
  if (w >= rows) return;
  const float* __restrict__ x = X + (size_t)w * ldx;
  float s = 0.f, s2 = 0.f;
  for (int c = lane; c < C; c += 32) { float v = x[c]; s += v; s2 += v * v; }
#pragma unroll
  for (int o = 16; o; o >>= 1) {
    s += __shfl_xor(s, o, 32);
    s2 += __shfl_xor(s2, o, 32);
  }
  const float mean = s / (float)C;
  const float var = s2 / (float)C - mean * mean;
  const float inv = rsqrtf(var + 1e-5f);
  float* __restrict__ y = Y + (size_t)w * ldy;
  for (int c = lane; c < C; c += 32) {
    float v = (x[c] - mean) * inv * g[c] + b[c];
    if (Z) {
      const float zv = Z[(size_t)w * ldz + c];
      v *= zv / (1.f + expf(-zv));            // silu gate
    }
    y[c] = v;
  }
}

// ------------------ depthwise 3x3 + SiLU, channel-last ---------------------
__global__ void dwconv3x3_silu(const float* __restrict__ X, int ldx,
                               const float* __restrict__ wgt,
                               const float* __restrict__ bias,
                               float* __restrict__ Y, int ldy,
                               int H, int W, int di) {
  const int d = blockIdx.x * blockDim.x + threadIdx.x;
  if (d >= di) return;
  const int l = blockIdx.y;
  const int b = blockIdx.z;
  const int h = l / W, wc = l - h * W;
  float acc = bias[d];
#pragma unroll
  for (int i = -1; i <= 1; ++i)
#pragma unroll
    for (int j = -1; j <= 1; ++j) {
      const int hh = h + i, ww = wc + j;
      if (hh >= 0 && hh < H && ww >= 0 && ww < W)
        acc += wgt[d * 9 + (i + 1) * 3 + (j + 1)] *
               X[((size_t)(b * H + hh) * W + ww) * ldx + d];
    }
  Y[((size_t)b * H * W + l) * ldy + d] = acc / (1.f + expf(-acc));
}

// -------- build 4 directional sequences xs (K=4, B, L, di) ----------------
__global__ void build_xs(const float* __restrict__ XM, float* __restrict__ XS,
                         int H, int W, int di) {
  const int d = blockIdx.x * blockDim.x + threadIdx.x;
  if (d >= di) return;
  const int l = blockIdx.y;
  const int b = blockIdx.z;
  const int L = H * W;
  const float v = XM[((size_t)b * L + l) * di + d];
  const int h = l / W, wc = l - h * W;
  const int lt = wc * H + h;                   // W/H transposed scan order
  const size_t SK = (size_t)BATCH * L * di;
  const size_t base = (size_t)b * L * di;
  XS[0 * SK + base + (size_t)l * di + d] = v;
  XS[1 * SK + base + (size_t)lt * di + d] = v;
  XS[2 * SK + base + (size_t)(L - 1 - l) * di + d] = v;
  XS[3 * SK + base + (size_t)(L - 1 - lt) * di + d] = v;
}

// ----------------- delta = softplus(delta_pre + dt_b) ----------------------
__global__ void softplus_bias(float* __restrict__ Dlt,
                              const float* __restrict__ dtb, int di) {
  const size_t total = (size_t)4 * BL * di;
  const size_t i = (size_t)blockIdx.x * blockDim.x + threadIdx.x;
  if (i >= total) return;
  const int d = (int)(i % di);
  const int k = (int)(i / ((size_t)BL * di));
  const float x = Dlt[i] + dtb[k * di + d];
  Dlt[i] = (x > 20.f) ? x : log1pf(expf(x));
}

// --------------------------- selective scan --------------------------------
// Thread = one (k,b,d) channel; h[16] state in VGPRs; B_t/C_t addresses are
// wave-uniform (k,b,l) -> scalarizable; u/dt loads coalesced over d.
__global__ void selective_scan(const float* __restrict__ XS,
                               const float* __restrict__ Dlt,
                               const float* __restrict__ XDBL,  // ld = C2P
                               const float* __restrict__ A_log,
                               const float* __restrict__ Dp,
                               float* __restrict__ YS,
                               int L, int di, int dtr) {
  const int d = blockIdx.x * blockDim.x + threadIdx.x;
  if (d >= di) return;
  const int b = blockIdx.y;
  const int k = blockIdx.z;

  float Av[NSTATE];
#pragma unroll
  for (int n = 0; n < NSTATE; ++n)
    Av[n] = -expf(A_log[((size_t)k * di + d) * NSTATE + n]);
  const float Dv = Dp[k * di + d];

  float h[NSTATE];
#pragma unroll
  for (int n = 0; n < NSTATE; ++n) h[n] = 0.f;

  const size_t row = ((size_t)k * BATCH + b) * L;
  const float* __restrict__ xs = XS + row * di + d;
  const float* __restrict__ dl = Dlt + row * di + d;
  const float* __restrict__ xd = XDBL + row * C2P;
  float* __restrict__ ys = YS + row * di + d;

  for (int l = 0; l < L; ++l) {
    const float u = xs[(size_t)l * di];
    const float dt = dl[(size_t)l * di];
    const float* __restrict__ Brow = xd + (size_t)l * C2P + dtr;
    const float* __restrict__ Crow = Brow + NSTATE;
    float y = 0.f;
#pragma unroll
    for (int n = 0; n < NSTATE; ++n) {
      const float dA = __expf(dt * Av[n]);
      h[n] = dA * h[n] + dt * Brow[n] * u;
      y += h[n] * Crow[n];
    }
    ys[(size_t)l * di] = y + u * Dv;
  }
}

// ------------- combine 4 directional outputs into (B,L,di) -----------------
__global__ void combine_y(const float* __restrict__ YS, float* __restrict__ YC,
                          int H, int W, int di) {
  const int d = blockIdx.x * blockDim.x + threadIdx.x;
  if (d >= di) return;
  const int l = blockIdx.y;
  const int b = blockIdx.z;
  const int L = H * W;
  const int h = l / W, wc = l - h * W;
  const int lt = wc * H + h;
  const size_t SK = (size_t)BATCH * L * di;
  const size_t base = (size_t)b * L * di;
  const float v = YS[0 * SK + base + (size_t)l * di + d] +
                  YS[2 * SK + base + (size_t)(L - 1 - l) * di + d] +
                  YS[1 * SK + base + (size_t)lt * di + d] +
                  YS[3 * SK + base + (size_t)(L - 1 - lt) * di + d];
  YC[base + (size_t)l * di + d] = v;
}

// ------------------------- weight padding kernels --------------------------
// x_proj_w (4, C2, di) -> (4, C2P, di), zero rows C2..C2P-1
__global__ void pad_xproj(const float* __restrict__ src, float* __restrict__ dst,
                          int C2, int di) {
  const size_t total = (size_t)4 * C2P * di;
  const size_t i = (size_t)blockIdx.x * blockDim.x + threadIdx.x;
  if (i >= total) return;
  const int d = (int)(i % di);
  const size_t r = i / di;
  const int c = (int)(r % C2P);
  const int k = (int)(r / C2P);
  dst[i] = (c < C2) ? src[((size_t)k * C2 + c) * di + d] : 0.f;
}
// dt_w (4, di, dtr) -> (4, di, dtrp), zero cols dtr..dtrp-1
__global__ void pad_dtw(const float* __restrict__ src, float* __restrict__ dst,
                        int dtr, int dtrp, int di) {
  const size_t total = (size_t)4 * di * dtrp;
  const size_t i = (size_t)blockIdx.x * blockDim.x + threadIdx.x;
  if (i >= total) return;
  const int c = (int)(i % dtrp);
  const size_t r = i / dtrp;
  const int d = (int)(r % di);
  const int k = (int)(r / di);
  dst[i] = (c < dtr) ? src[((size_t)k * di + d) * dtr + c] : 0.f;
}

// ------------------------- misc small kernels ------------------------------
__global__ void nchw_to_nlc(const float* __restrict__ in, float* __restrict__ out,
                            int C, int L) {
  const size_t i = (size_t)blockIdx.x * blockDim.x + threadIdx.x;
  const size_t total = (size_t)BATCH * L * C;
  if (i >= total) return;
  const int c = (int)(i % C);
  const size_t r = i / C;
  const int l = (int)(r % L);
  const int b = (int)(r / L);
  out[i] = in[((size_t)b * C + c) * L + l];
}

__global__ void copy_cols(const float* __restrict__ S, int lds,
                          float* __restrict__ Dst, int ldd,
                          int rows, int C, int colOff) {
  const size_t i = (size_t)blockIdx.x * blockDim.x + threadIdx.x;
  if (i >= (size_t)rows * C) return;
  const int c = (int)(i % C);
  const size_t r = i / C;
  Dst[r * ldd + colOff + c] = S[r * lds + c];
}

__global__ void bn_relu_nchw(const float* __restrict__ Ycl,
                             const float* __restrict__ g,
                             const float* __restrict__ bb,
                             const float* __restrict__ mn,
                             const float* __restrict__ vr,
                             float* __restrict__ out, int L, int O) {
  const size_t i = (size_t)blockIdx.x * blockDim.x + threadIdx.x;
  const size_t total = (size_t)BATCH * O * L;
  if (i >= total) return;
  const int l = (int)(i % L);
  const size_t r = i / L;
  const int o = (int)(r % O);
  const int b = (int)(r / O);
  const float x = Ycl[((size_t)b * L + l) * O + o];
  const float y = (x - mn[o]) * rsqrtf(vr[o] + 1e-5f) * g[o] + bb[o];
  out[i] = fmaxf(y, 0.f);
}

// ---------------------------- host driver ----------------------------------
static inline void launch_gemm(hipStream_t s, const float* A, int lda,
                               const float* Bw, int ldb, float* D, int ldd,
                               const float* bias, const float* add, int ldadd,
                               int M, int N, int K, int relu = 0) {
  const int tiles = (M / MTILE) * (N / 16);
  const int wpb = 4;  // waves per block
  const int grid = (tiles + wpb - 1) / wpb;
  wmma_gemm_nt<<<grid, 32 * wpb, 0, s>>>(A, lda, Bw, ldb, D, ldd, bias, add,
                                         ldadd, M, N, K, relu);
}

static inline void launch_ln(hipStream_t s, const float* X, int ldx,
                             const float* g, const float* b, const float* Z,
                             int ldz, float* Y, int ldy, int rows, int C) {
  const int threads = 256;                       // 8 waves per block
  const int grid = (rows * 32 + threads - 1) / threads;
  layernorm_gate<<<grid, threads, 0, s>>>(X, ldx, g, b, Z, ldz, Y, ldy, rows, C);
}

extern "C" void kernel_launch(void* const* d_in, const int* in_sizes, int n_in,
                              void* d_out, int out_size, void* d_ws, size_t ws_size,
                              hipStream_t stream) {
  (void)in_sizes; (void)n_in; (void)out_size; (void)ws_size;

  // ---- workspace carve (fp32), sized for the largest stage (s4) ----
  float* ws = (float*)d_ws;
  size_t off = 0;
  auto alloc = [&](size_t n) { float* p = ws + off; off += n; return p; };
  float* bufA = alloc((size_t)BL * 256);
  float* bufB = alloc((size_t)BL * 256);
  float* y1   = alloc((size_t)BL * 256);
  float* xln  = alloc((size_t)BL * 256);
  float* xz   = alloc((size_t)BL * 1024);
  float* xm   = alloc((size_t)BL * 512);
  float* xs   = alloc((size_t)4 * BL * 512);
  float* xdbl = alloc((size_t)4 * BL * C2P);
  float* dlt  = alloc((size_t)4 * BL * 512);
  float* ysb  = alloc((size_t)4 * BL * 512);
  float* ycmb = alloc((size_t)BL * 512);
  float* ygt  = alloc((size_t)BL * 512);
  float* mcat = alloc((size_t)BL * 480);
  float* yf   = alloc((size_t)BL * 1024);
  float* xpw  = alloc((size_t)4 * C2P * 512);   // padded x_proj weights
  float* dtwp = alloc((size_t)4 * 512 * 16);    // padded dt weights

  const float* x_in = (const float*)d_in[0];

  // initial NCHW (4,256,32,32) -> channel-last (4096, 256)
  {
    const size_t total = (size_t)BL * 256;
    nchw_to_nlc<<<(int)((total + 255) / 256), 256, 0, stream>>>(x_in, bufA, 256, LL);
  }

  const int cins[4]  = {256, 32, 64, 128};
  const int couts[4] = {32, 64, 128, 256};
  const int colOffs[4] = {0, 32, 96, 224};

  float* cur = bufA;
  float* nxt = bufB;

  for (int s = 0; s < 4; ++s) {
    const int base = 1 + s * 15;
    const float* conv1_w   = (const float*)d_in[base + 0];
    const float* conv1_b   = (const float*)d_in[base + 1];
    const float* ln_g      = (const float*)d_in[base + 2];
    const float* ln_b      = (const float*)d_in[base + 3];
    const float* in_proj_w = (const float*)d_in[base + 4];
    const float* conv_w    = (const float*)d_in[base + 5];
    const float* conv_b    = (const float*)d_in[base + 6];
    const float* x_proj_w  = (const float*)d_in[base + 7];
    const float* dt_w      = (const float*)d_in[base + 8];
    const float* dt_b      = (const float*)d_in[base + 9];
    const float* A_log     = (const float*)d_in[base + 10];
    const float* Dp        = (const float*)d_in[base + 11];
    const float* out_ln_g  = (const float*)d_in[base + 12];
    const float* out_ln_b  = (const float*)d_in[base + 13];
    const float* out_proj_w= (const float*)d_in[base + 14];

    const int cin = cins[s], cout = couts[s];
    const int di = 2 * cout;
    const int dtr = cout / 16;                 // 2,4,8,16
    const int dtrp = (dtr < 4) ? 4 : dtr;      // K padded to multiple of 4
    const int C2 = dtr + 2 * NSTATE;           // 34,36,40,48

    // 0) pad small weights -> branch-free GEMMs
    {
      const size_t t1 = (size_t)4 * C2P * di;
      pad_xproj<<<(int)((t1 + 255) / 256), 256, 0, stream>>>(x_proj_w, xpw, C2, di);
      const size_t t2 = (size_t)4 * di * dtrp;
      pad_dtw<<<(int)((t2 + 255) / 256), 256, 0, stream>>>(dt_w, dtwp, dtr, dtrp, di);
    }
    // 1) conv1x1 (channel-last NT GEMM)
    launch_gemm(stream, cur, cin, conv1_w, cin, y1, cout, conv1_b, nullptr, 0,
                BL, cout, cin);
    // 2) LayerNorm
    launch_ln(stream, y1, cout, ln_g, ln_b, nullptr, 0, xln, cout, BL, cout);
    // 3) in_proj -> xz (BL x 2di): xm = xz[:, :di], z = xz[:, di:]
    launch_gemm(stream, xln, cout, in_proj_w, cout, xz, 2 * di, nullptr,
                nullptr, 0, BL, 2 * di, cout);
    // 4) depthwise 3x3 + SiLU (read xm half of xz with ld = 2di)
    {
      dim3 g((di + 127) / 128, LL, BATCH);
      dwconv3x3_silu<<<g, 128, 0, stream>>>(xz, 2 * di, conv_w, conv_b, xm, di,
                                            HH, WW, di);
    }
    // 5) four directional sequences
    {
      dim3 g((di + 127) / 128, LL, BATCH);
      build_xs<<<g, 128, 0, stream>>>(xm, xs, HH, WW, di);
    }
    // 6) x_proj per direction: (BL x di) @ (C2P x di)^T -> (BL x C2P)
    //    padded weight rows are zero -> xdbl cols C2..C2P-1 are exactly 0
    for (int k = 0; k < 4; ++k)
      launch_gemm(stream, xs + (size_t)k * BL * di, di,
                  xpw + (size_t)k * C2P * di, di,
                  xdbl + (size_t)k * BL * C2P, C2P, nullptr, nullptr, 0,
                  BL, C2P, di);
    // 7) dt projection: K widened to dtrp; padded zero weights cancel the
    //    Bs columns the wider K-read touches
    for (int k = 0; k < 4; ++k)
      launch_gemm(stream, xdbl + (size_t)k * BL * C2P, C2P,
                  dtwp + (size_t)k * di * dtrp, dtrp,
                  dlt + (size_t)k * BL * di, di, nullptr, nullptr, 0,
                  BL, di, dtrp);
    // 8) delta = softplus(. + dt_b)
    {
      const size_t total = (size_t)4 * BL * di;
      softplus_bias<<<(int)((total + 255) / 256), 256, 0, stream>>>(dlt, dt_b, di);
    }
    // 9) selective scan
    {
      dim3 g(di / 64, BATCH, 4);
      selective_scan<<<g, 64, 0, stream>>>(xs, dlt, xdbl, A_log, Dp, ysb,
                                           LL, di, dtr);
    }
    // 10) combine 4 directions
    {
      dim3 g((di + 127) / 128, LL, BATCH);
      combine_y<<<g, 128, 0, stream>>>(ysb, ycmb, HH, WW, di);
    }
    // 11) out LN fused with silu(z) gate  (z = xz + di, ld 2di)
    launch_ln(stream, ycmb, di, out_ln_g, out_ln_b, xz + di, 2 * di, ygt, di,
              BL, di);
    // 12) out_proj + residual (y1): stage output channel-last (BL x cout)
    launch_gemm(stream, ygt, di, out_proj_w, di, nxt, cout, nullptr, y1, cout,
                BL, cout, di);
    // 13) scatter into concat buffer
    {
      const size_t total = (size_t)BL * cout;
      copy_cols<<<(int)((total + 255) / 256), 256, 0, stream>>>(
          nxt, cout, mcat, 480, BL, cout, colOffs[s]);
    }
    float* t = cur; cur = nxt; nxt = t;
  }

  // ---- final head: (BL x 480) @ (1024 x 480)^T + b, BN + ReLU, NCHW out ----
  const int fb = 61;
  const float* fw   = (const float*)d_in[fb + 0];
  const float* fbi  = (const float*)d_in[fb + 1];
  const float* bn_g = (const float*)d_in[fb + 2];
  const float* bn_b = (const float*)d_in[fb + 3];
  const float* bn_m = (const float*)d_in[fb + 4];
  const float* bn_v = (const float*)d_in[fb + 5];

  launch_gemm(stream, mcat, 480, fw, 480, yf, 1024, fbi, nullptr, 0,
              BL, 1024, 480);
  {
    const size_t total = (size_t)BATCH * 1024 * LL;
    bn_relu_nchw<<<(int)((total + 255) / 256), 256, 0, stream>>>(
        yf, bn_g, bn_b, bn_m, bn_v, (float*)d_out, LL, 1024);
  }
}